// DAIN_81054622810327
// MI455X (gfx1250) — compile-verified
//
#include <hip/hip_runtime.h>
#include <stdint.h>

#define BB 4
#define H 384
#define W 512
#define HW (H*W)
#define IH 96
#define IW 128
#define FSCALE 10.0f   // DIV_FLOW * T

typedef float v2f __attribute__((ext_vector_type(2)));
typedef float v8f __attribute__((ext_vector_type(8)));

__device__ __forceinline__ int iclamp(int v, int lo, int hi) {
    return v < lo ? lo : (v > hi ? hi : v);
}

// ---------------------------------------------------------------------------
// K1: bilinear x4 upsample along X via V_WMMA_F32_16X16X4_F32.
// in : [nch][IH][IW]   out : [nch][IH][W], values pre-scaled by FSCALE.
// Each 16-wide output chunk (x = 16t..16t+15) = Wmat(16x6) * in[4t-1..4t+4]
// (edge-clamped), done for 16 rows at once:  D(16x16) = A(16x8) x B(8x16),
// split into two K=4 WMMAs (taps 6,7 are zero padding).
// One wave (32 threads) per tile; EXEC all ones at the WMMA.
// ---------------------------------------------------------------------------
__global__ __launch_bounds__(32)
void upx_wmma(const float* __restrict__ in, float* __restrict__ out)
{
    const int tile = blockIdx.x;
    const int ct   = tile & 31;                 // W/16 = 32 column tiles
    const int rt   = (tile >> 5) % (IH / 16);   // 6 row tiles
    const int ch   = tile / (32 * (IH / 16));
    const int lane = threadIdx.x;
    const int half = lane >> 4;                 // K-half select (ISA layout)
    const int m    = lane & 15;                 // A row / B column / D column

    const float* src = in + (size_t)ch * IH * IW + (rt * 16 + m) * IW;
    const int xbase = 4 * ct - 1;

    // B matrix (8x16): VGPR0 holds K={0|2}(chunk0),{4|6}(chunk1) per lane half
    v2f b0, b1;
    b0.x = src[iclamp(xbase + 2 * half,     0, IW - 1)];
    b0.y = src[iclamp(xbase + 2 * half + 1, 0, IW - 1)];
    if (half == 0) {  // taps 4,5 valid; taps 6,7 (other half) are zero
        b1.x = src[iclamp(xbase + 4, 0, IW - 1)];
        b1.y = src[iclamp(xbase + 5, 0, IW - 1)];
    } else {
        b1.x = 0.0f; b1.y = 0.0f;
    }

    // A matrix (16x8): row m = output phase; src pos = m/4 - 0.375
    const int   q  = m >> 2, r = m & 3;
    const int   kA = q + (r >= 2 ? 1 : 0);
    const float wA = (r == 0) ? 0.375f : (r == 1) ? 0.125f
                   : (r == 2) ? 0.875f : 0.625f;
    auto wgt = [&](int k) -> float {
        float wk = (k == kA) ? wA : ((k == kA + 1) ? (1.0f - wA) : 0.0f);
        return wk * FSCALE;                      // fold flow scale into weights
    };
    v2f a0, a1;
    a0.x = wgt(2 * half);     a0.y = wgt(2 * half + 1);
    a1.x = wgt(4 + 2 * half); a1.y = wgt(5 + 2 * half);

    v8f d = {};
    d = __builtin_amdgcn_wmma_f32_16x16x4_f32(false, a0, false, b0,
                                              (short)0, d, false, false);
    d = __builtin_amdgcn_wmma_f32_16x16x4_f32(false, a1, false, b1,
                                              (short)0, d, false, false);

    // D: lane<16 -> (M=v,   N=lane);  lane>=16 -> (M=v+8, N=lane-16)
    // => each lane stores 8 consecutive x's of one row.
    float* dst = out + (size_t)ch * IH * W + (rt * 16 + m) * W + ct * 16 + half * 8;
    *(float4*)(dst)     = make_float4(d[0], d[1], d[2], d[3]);
    *(float4*)(dst + 4) = make_float4(d[4], d[5], d[6], d[7]);
}

// ---------------------------------------------------------------------------
// K2: bilinear x4 upsample along Y (2-tap lerp, coalesced along x).
// tmpx: [nch][IH][W] -> out: [nch][H][W]
// ---------------------------------------------------------------------------
__global__ void upy(const float* __restrict__ tmpx, float* __restrict__ out, int total)
{
    int i = blockIdx.x * blockDim.x + threadIdx.x;
    if (i >= total) return;
    int x  = i % W;
    int yo = (i / W) % H;
    int ch = i / (W * H);
    int g = yo >> 2, r = yo & 3;
    int yA = (r < 2) ? g - 1 : g;
    float wA = (r == 0) ? 0.375f : (r == 1) ? 0.125f
             : (r == 2) ? 0.875f : 0.625f;
    int y0 = iclamp(yA, 0, IH - 1), y1 = iclamp(yA + 1, 0, IH - 1);
    const float* b = tmpx + (size_t)ch * IH * W;
    out[i] = wA * b[y0 * W + x] + (1.0f - wA) * b[y1 * W + x];
}

// ---------------------------------------------------------------------------
// K3: FlowProjection scatter: acc[3] = {accx, accy, cnt} per batch.
// ---------------------------------------------------------------------------
__global__ void fproj_scatter(const float* __restrict__ fup, float* __restrict__ acc, int total)
{
    int i = blockIdx.x * blockDim.x + threadIdx.x;
    if (i >= total) return;
    int x = i % W, y = (i / W) % H, b = i / HW;
    float fx = fup[(size_t)(b * 2 + 0) * HW + y * W + x];
    float fy = fup[(size_t)(b * 2 + 1) * HW + y * W + x];
    float x2 = (float)x + fx, y2 = (float)y + fy;
    if (!(x2 >= 0.0f && x2 <= (float)(W - 1) && y2 >= 0.0f && y2 <= (float)(H - 1)))
        return;  // invalid lanes contribute exactly zero in the reference
    int xf = (int)floorf(x2), yf = (int)floorf(y2);
    float* ax = acc + (size_t)(b * 3 + 0) * HW;
    float* ay = acc + (size_t)(b * 3 + 1) * HW;
    float* ac = acc + (size_t)(b * 3 + 2) * HW;
#pragma unroll
    for (int dy = 0; dy < 2; ++dy)
#pragma unroll
        for (int dx = 0; dx < 2; ++dx) {
            int idx = iclamp(yf + dy, 0, H - 1) * W + iclamp(xf + dx, 0, W - 1);
            atomicAdd(ax + idx, -fx);
            atomicAdd(ay + idx, -fy);
            atomicAdd(ac + idx, 1.0f);
        }
}

// K4: average by hit count; write avg into fup (flow buffer is now free).
__global__ void fproj_avg(float* __restrict__ fup, const float* __restrict__ acc, int total)
{
    int i = blockIdx.x * blockDim.x + threadIdx.x;
    if (i >= total) return;
    int p = i % HW, b = i / HW;
    float dnm = fmaxf(acc[(size_t)(b * 3 + 2) * HW + p], 1.0f);
    fup[(size_t)(b * 2 + 0) * HW + p] = acc[(size_t)(b * 3 + 0) * HW + p] / dnm;
    fup[(size_t)(b * 2 + 1) * HW + p] = acc[(size_t)(b * 3 + 1) * HW + p] / dnm;
}

// K5: hole fill with valid 4-neighbor mean; final flow into acc[0..1].
__global__ void fproj_fill(const float* __restrict__ fup, float* __restrict__ acc, int total)
{
    int i = blockIdx.x * blockDim.x + threadIdx.x;
    if (i >= total) return;
    int x = i % W, y = (i / W) % H, b = i / HW;
    const float* cnt = acc + (size_t)(b * 3 + 2) * HW;
    const float* ax  = fup + (size_t)(b * 2 + 0) * HW;
    const float* ay  = fup + (size_t)(b * 2 + 1) * HW;
    int p = y * W + x;
    float c = cnt[p], vx = ax[p], vy = ay[p];
    float numx = 0.f, numy = 0.f, den = 0.f;
    const int dxs[4] = {0, 0, -1, 1}, dys[4] = {-1, 1, 0, 0};
#pragma unroll
    for (int t = 0; t < 4; ++t) {
        int xn = x + dxs[t], yn = y + dys[t];
        if (xn >= 0 && xn < W && yn >= 0 && yn < H) {
            int pn = yn * W + xn;
            if (cnt[pn] > 0.0f) { numx += ax[pn]; numy += ay[pn]; den += 1.0f; }
        }
    }
    float dd = fmaxf(den, 1.0f);
    acc[(size_t)(b * 3 + 0) * HW + p] = (c <= 0.0f) ? numx / dd : vx;
    acc[(size_t)(b * 3 + 1) * HW + p] = (c <= 0.0f) ? numy / dd : vy;
}

// ---------------------------------------------------------------------------
// K6: fused FilterInterpolation(img0,Ft0,filt0) + (img2,Ft2,filt1), blended.
// 4x4 taps share bilinear corners -> gather 5x5 per channel (25 loads vs 64).
// The 32 filter coefficients per pixel are staged to LDS with
// global_load_async_to_lds_b32 (ASYNCcnt) and read back via ds_load.
// ---------------------------------------------------------------------------
__global__ __launch_bounds__(128)
void interp_blend(const float* __restrict__ img0, const float* __restrict__ img2,
                  const float* __restrict__ ft0,  const float* __restrict__ ft2,
                  const float* __restrict__ filt0, const float* __restrict__ filt1,
                  float* __restrict__ out)
{
    __shared__ float sfilt[2][16][128];
    const int tx = threadIdx.x;
    const int blk = blockIdx.x;
    const int xb = blk & 3;                 // W / 128 = 4 tiles per row
    const int y  = (blk >> 2) % H;
    const int b  = blk / (4 * H);
    const int x  = xb * 128 + tx;
    const size_t pix = (size_t)y * W + x;

    // Stage filt0/filt1 rows for this pixel strip: 32 async copies per lane.
    const float* fsrc[2] = { filt0, filt1 };
    uint32_t ldsbase = (uint32_t)(uintptr_t)(&sfilt[0][0][0]);
#pragma unroll
    for (int im = 0; im < 2; ++im) {
        const float* fp = fsrc[im] + ((size_t)b * 16) * HW + pix;
#pragma unroll
        for (int k = 0; k < 16; ++k) {
            uint64_t ga = (uint64_t)(uintptr_t)(fp + (size_t)k * HW);
            uint32_t la = ldsbase + (uint32_t)(((im * 16 + k) * 128 + tx) * 4);
            asm volatile("global_load_async_to_lds_b32 %0, %1, off"
                         :: "v"(la), "v"(ga) : "memory");
        }
    }

    const float* imgs[2] = { img0, img2 };
    const float* fts[2]  = { ft0,  ft2 };
    float o[3] = {0.0f, 0.0f, 0.0f};

    asm volatile("s_wait_asynccnt 0x0" ::: "memory");

#pragma unroll
    for (int im = 0; im < 2; ++im) {
        float Fx = fts[im][(size_t)(b * 3 + 0) * HW + pix];
        float Fy = fts[im][(size_t)(b * 3 + 1) * HW + pix];
        float x2 = fminf(fmaxf((float)x + Fx, 0.0f), (float)(W - 1));
        float y2 = fminf(fmaxf((float)y + Fy, 0.0f), (float)(H - 1));
        float xff = floorf(x2), yff = floorf(y2);
        int xf = (int)xff, yf = (int)yff;
        float a  = x2 - xff, bb = y2 - yff;
        float w00 = (1.0f - a) * (1.0f - bb), w10 = a * (1.0f - bb);
        float w01 = (1.0f - a) * bb,          w11 = a * bb;
        int xc[5], yc[5];
#pragma unroll
        for (int t = 0; t < 5; ++t) {
            xc[t] = iclamp(xf - 1 + t, 0, W - 1);
            yc[t] = iclamp(yf - 1 + t, 0, H - 1);
        }
        float f[16];
#pragma unroll
        for (int k = 0; k < 16; ++k) f[k] = sfilt[im][k][tx];
#pragma unroll
        for (int c = 0; c < 3; ++c) {
            const float* base = imgs[im] + (size_t)(b * 3 + c) * HW;
            float I[5][5];
#pragma unroll
            for (int j = 0; j < 5; ++j)
#pragma unroll
                for (int ii = 0; ii < 5; ++ii)
                    I[j][ii] = __ldg(base + (size_t)yc[j] * W + xc[ii]);
            float acc = 0.0f;
#pragma unroll
            for (int j = 0; j < 4; ++j)
#pragma unroll
                for (int ii = 0; ii < 4; ++ii) {
                    float v = w00 * I[j][ii]     + w10 * I[j][ii + 1]
                            + w01 * I[j + 1][ii] + w11 * I[j + 1][ii + 1];
                    acc = fmaf(f[j * 4 + ii], v, acc);
                }
            o[c] += acc;
        }
    }
#pragma unroll
    for (int c = 0; c < 3; ++c)
        out[(size_t)(b * 3 + c) * HW + pix] = 0.5f * o[c];
}

// ---------------------------------------------------------------------------
extern "C" void kernel_launch(void* const* d_in, const int* in_sizes, int n_in,
                              void* d_out, int out_size, void* d_ws, size_t ws_size,
                              hipStream_t stream)
{
    const float* input0 = (const float*)d_in[0];
    const float* input2 = (const float*)d_in[1];
    const float* flow01 = (const float*)d_in[2];
    const float* flow10 = (const float*)d_in[3];
    const float* filt0  = (const float*)d_in[4];
    const float* filt1  = (const float*)d_in[5];
    float* out = (float*)d_out;

    float* w    = (float*)d_ws;
    float* tmpx = w;                               // B*2 * IH * W
    float* fup0 = tmpx + (size_t)BB * 2 * IH * W;  // B*2 * H * W
    float* fup1 = fup0 + (size_t)BB * 2 * HW;
    float* acc0 = fup1 + (size_t)BB * 2 * HW;      // B*3 * H * W
    float* acc1 = acc0 + (size_t)BB * 3 * HW;

    const int upx_grid = BB * 2 * (IH / 16) * (W / 16);
    const int nup = BB * 2 * HW;
    const int n   = BB * HW;

    hipMemsetAsync(acc0, 0, (size_t)BB * 3 * HW * sizeof(float), stream);
    hipMemsetAsync(acc1, 0, (size_t)BB * 3 * HW * sizeof(float), stream);

    // F_{0->2}*scale upsampled
    upx_wmma<<<upx_grid, 32, 0, stream>>>(flow01, tmpx);
    upy<<<(nup + 255) / 256, 256, 0, stream>>>(tmpx, fup0, nup);
    // F_{2->0}*scale upsampled
    upx_wmma<<<upx_grid, 32, 0, stream>>>(flow10, tmpx);
    upy<<<(nup + 255) / 256, 256, 0, stream>>>(tmpx, fup1, nup);

    fproj_scatter<<<(n + 255) / 256, 256, 0, stream>>>(fup0, acc0, n);
    fproj_scatter<<<(n + 255) / 256, 256, 0, stream>>>(fup1, acc1, n);
    fproj_avg<<<(n + 255) / 256, 256, 0, stream>>>(fup0, acc0, n);
    fproj_avg<<<(n + 255) / 256, 256, 0, stream>>>(fup1, acc1, n);
    fproj_fill<<<(n + 255) / 256, 256, 0, stream>>>(fup0, acc0, n);
    fproj_fill<<<(n + 255) / 256, 256, 0, stream>>>(fup1, acc1, n);

    interp_blend<<<BB * H * (W / 128), 128, 0, stream>>>(
        input0, input2, acc0, acc1, filt0, filt1, out);
}